// BaseSampler_19043884990816
// MI455X (gfx1250) — compile-verified
//
#include <hip/hip_runtime.h>
#include <hip/hip_bf16.h>

// ---------------------------------------------------------------------------
// MI455X (gfx1250) sampler:
//  Phase 1: one workgroup per candidate token. Async-DMA the 32000-float
//           logits row into LDS (global_load_async_to_lds_b128, ASYNCcnt),
//           then do max/argmax, exact top-k threshold via 4-pass radix select
//           (LDS histogram), survivor compaction + sort, nucleus (top-p)
//           prefix, score = 1/sum(exp) over kept set.
//  Phase 2: one wave32 per batch group: rank-sort <=32 scores, apply
//           rank/threshold filters, emit pos/tokens/counts.
// Roofline: 1056 * 32000 * 4B = 135 MB single HBM pass ~= 5.8 us @ 23.3 TB/s;
// all selection passes run out of LDS (320 KB/WGP holds the full row).
// No matmul in this op -> WMMA not applicable; CDNA5 path used is async
// global->LDS + split wait counters.
// ---------------------------------------------------------------------------

#define V_MAX   32000
#define NTH1    512

// dynamic-LDS layout (bytes)
#define OFF_ROW   0
#define OFF_REDV  (V_MAX * 4)             // float[NTH1]
#define OFF_REDI  (OFF_REDV + NTH1 * 4)   // int[NTH1]
#define OFF_HIST  (OFF_REDI + NTH1 * 4)   // int[256]
#define OFF_SURV  (OFF_HIST + 256 * 4)    // float[128]
#define OFF_SORT  (OFF_SURV + 128 * 4)    // float[128]
#define OFF_BC    (OFF_SORT + 128 * 4)    // int[16]
#define SMEM1     (OFF_BC + 64)

// monotonic float -> uint key (descending float == descending key)
__device__ __forceinline__ unsigned fkey(float x) {
  unsigned u = __float_as_uint(x);
  return (u & 0x80000000u) ? ~u : (u | 0x80000000u);
}

__global__ __launch_bounds__(NTH1)
void sampler_phase1(const float* __restrict__ logits,
                    const int*   __restrict__ cu_q,
                    const int*   __restrict__ rel,
                    const int*   __restrict__ boff,
                    const int*   __restrict__ cuf,
                    const float* __restrict__ temps,
                    const int*   __restrict__ topk,
                    const float* __restrict__ topp,
                    float* __restrict__ ws_score,
                    int*   __restrict__ ws_tok,
                    int*   __restrict__ ws_pos,
                    int B, int logits_numel) {
  extern __shared__ unsigned char smem[];
  float* rowv  = (float*)(smem + OFF_ROW);
  float* redv  = (float*)(smem + OFF_REDV);
  int*   redi  = (int*)  (smem + OFF_REDI);
  int*   hist  = (int*)  (smem + OFF_HIST);
  float* surv  = (float*)(smem + OFF_SURV);
  float* sortv = (float*)(smem + OFF_SORT);
  int*   bc    = (int*)  (smem + OFF_BC);

  const int t   = blockIdx.x;
  const int tid = threadIdx.x;

  // group id: cuf[g] <= t < cuf[g+1]   (uniform -> scalar loads)
  int g = 0;
  for (int i = 0; i < B; ++i) { if (t >= cuf[i + 1]) g = i + 1; }

  const int rows = cu_q[B];
  int V = (rows > 0) ? (logits_numel / rows) : V_MAX;
  if (V > V_MAX) V = V_MAX;

  const int row = cu_q[g] + rel[t];
  const float* gp = logits + (long long)row * (long long)V;

  // ---- async DMA: global row -> LDS (16B chunks, ASYNCcnt tracked) ----
  const int nchunk = V >> 2;
  for (int c = tid; c < nchunk; c += NTH1) {
    unsigned          ldsoff = (unsigned)(unsigned long long)(size_t)(rowv + (c << 2));
    unsigned long long ga    = (unsigned long long)(size_t)(gp + (c << 2));
    asm volatile("global_load_async_to_lds_b128 %0, %1, off"
                 :: "v"(ldsoff), "v"(ga) : "memory");
  }
  for (int j = (nchunk << 2) + tid; j < V; j += NTH1) rowv[j] = gp[j]; // tail
  asm volatile("s_wait_asynccnt 0" ::: "memory");
  __syncthreads();

  // ---- max / argmax (first-occurrence tie break) ----
  float bv = -__builtin_inff();
  int   bi = 0x7fffffff;
  for (int i = tid; i < V; i += NTH1) {
    float v = rowv[i];
    if (v > bv) { bv = v; bi = i; }
  }
  redv[tid] = bv; redi[tid] = bi;
  for (int s = NTH1 / 2; s > 0; s >>= 1) {
    __syncthreads();
    if (tid < s) {
      float ov = redv[tid + s]; int oi = redi[tid + s];
      float mv = redv[tid];     int mi = redi[tid];
      if (ov > mv || (ov == mv && oi < mi)) { redv[tid] = ov; redi[tid] = oi; }
    }
  }
  __syncthreads();
  const float maxv = redv[0];
  const int   maxi = redi[0];

  // ---- exact k-th largest via MSB-first radix select (LDS histogram) ----
  int k = topk[t];
  if (k < 1) k = 1;
  if (k > 64) k = 64;
  if (k > V) k = V;

  unsigned prefix = 0, pmask = 0;
  int rem = k;
  for (int shift = 24; shift >= 0; shift -= 8) {
    for (int i = tid; i < 256; i += NTH1) hist[i] = 0;
    __syncthreads();
    for (int i = tid; i < V; i += NTH1) {
      unsigned kk = fkey(rowv[i]);
      if ((kk & pmask) == prefix) atomicAdd(&hist[(kk >> shift) & 0xFFu], 1);
    }
    __syncthreads();
    if (tid == 0) {
      int c = 0, sel = 0, r2 = rem;
      for (int bin = 255; bin >= 0; --bin) {
        int h = hist[bin];
        if (c + h >= rem) { sel = bin; r2 = rem - c; break; }
        c += h;
      }
      bc[0] = (int)(prefix | ((unsigned)sel << shift));
      bc[1] = r2;
    }
    __syncthreads();
    prefix = (unsigned)bc[0];
    rem    = bc[1];
    pmask |= (0xFFu << shift);
  }
  const unsigned thrkey = prefix;   // key of exact k-th largest element

  // ---- compact survivors (value >= threshold; includes ties) ----
  if (tid == 0) bc[2] = 0;
  __syncthreads();
  for (int i = tid; i < V; i += NTH1) {
    float v = rowv[i];
    if (fkey(v) >= thrkey) {
      int p = atomicAdd(&bc[2], 1);
      if (p < 128) surv[p] = v;
    }
  }
  __syncthreads();
  int n = bc[2]; if (n > 128) n = 128;

  // ---- rank sort survivors descending (stable) ----
  if (tid < n) {
    float mv = surv[tid];
    unsigned mk = fkey(mv);
    int rank = 0;
    for (int j = 0; j < n; ++j) {
      unsigned kj = fkey(surv[j]);
      rank += (kj > mk) || (kj == mk && j < tid);
    }
    sortv[rank] = mv;
  }
  __syncthreads();

  // ---- nucleus prefix + score (serial over n<=128) ----
  if (tid == 0) {
    const float invT = 1.0f / temps[t];
    const float m = maxv * invT;
    float denom_all = 0.0f;
    for (int i = 0; i < n; ++i) {
      float e = expf(sortv[i] * invT - m);
      surv[i] = e;                 // reuse as exp buffer
      denom_all += e;
    }
    const float p = topp[t];
    float cum = 0.0f, denomK = 0.0f;
    for (int i = 0; i < n; ++i) {
      float e = surv[i];
      if (i == 0 || cum <= p) denomK += e;   // keep = prefix where prev-cum <= p
      cum += e / denom_all;
    }
    ws_score[t] = 1.0f / denomK;   // max prob: e0 = exp(0) = 1
    ws_tok[t]   = maxi;
    ws_pos[t]   = rel[t] + boff[g];
  }
}

__global__ __launch_bounds__(32)
void sampler_phase2(const int*   __restrict__ cuf,
                    const int*   __restrict__ ntr,
                    const float* __restrict__ thrs,
                    const int*   __restrict__ bsp,
                    const float* __restrict__ wsc,
                    const int*   __restrict__ wtk,
                    const int*   __restrict__ wps,
                    int* __restrict__ out, int B) {
  __shared__ float sc[32];
  __shared__ int   tk[32];
  __shared__ int   ps[32];
  __shared__ int   cnt;

  const int b = blockIdx.x;
  const int lane = threadIdx.x;
  const int bs = *bsp;

  const int st = cuf[b];
  const int n  = cuf[b + 1] - st;

  float s = -__builtin_inff();
  int tok = 0, pos = 0;
  if (lane < n) { s = wsc[st + lane]; tok = wtk[st + lane]; pos = wps[st + lane]; }
  sc[lane] = s; tk[lane] = tok; ps[lane] = pos;
  if (lane == 0) cnt = 0;
  __syncthreads();

  // stable descending rank among 32 slots
  int rank = 0;
  for (int j = 0; j < 32; ++j) {
    float sj = sc[j];
    rank += (sj > s) || (sj == s && j < lane);
  }

  int kv = ntr[b]; if (kv < 0) kv = 0;
  const float th = thrs[b];
  const bool keep = (rank < kv) && (s >= th) && (s > -__builtin_inff());

  int* out_pos = out;
  int* out_tok = out + B * bs;
  int* out_cnt = out + 2 * B * bs;

  if (rank < bs) {
    out_pos[b * bs + rank] = keep ? pos : 0;
    out_tok[b * bs + rank] = keep ? tok : -1;
  }
  for (int i = 32 + lane; i < bs; i += 32) {   // any slots beyond 32 are empty
    out_pos[b * bs + i] = 0;
    out_tok[b * bs + i] = -1;
  }
  if (keep) atomicAdd(&cnt, 1);
  __syncthreads();
  if (lane == 0) out_cnt[b] = cnt;
}

extern "C" void kernel_launch(void* const* d_in, const int* in_sizes, int n_in,
                              void* d_out, int out_size, void* d_ws, size_t ws_size,
                              hipStream_t stream) {
  const float* logits = (const float*)d_in[0];
  const int*   cu_q   = (const int*)  d_in[1];
  const int*   rel    = (const int*)  d_in[2];
  const int*   boff   = (const int*)  d_in[3];
  const int*   cuf    = (const int*)  d_in[4];
  const float* temps  = (const float*)d_in[5];
  const int*   ntr    = (const int*)  d_in[6];
  const float* thrs   = (const float*)d_in[7];
  const int*   topk   = (const int*)  d_in[8];
  const float* topp   = (const float*)d_in[9];
  const int*   bsp    = (const int*)  d_in[10];

  const int T = in_sizes[2];          // ragged candidate count
  const int B = in_sizes[1] - 1;      // batch groups
  const int logits_numel = in_sizes[0];

  float* ws_score = (float*)d_ws;
  int*   ws_tok   = (int*)d_ws + T;
  int*   ws_pos   = (int*)d_ws + 2 * T;

  sampler_phase1<<<T, NTH1, SMEM1, stream>>>(
      logits, cu_q, rel, boff, cuf, temps, topk, topp,
      ws_score, ws_tok, ws_pos, B, logits_numel);

  sampler_phase2<<<B, 32, 0, stream>>>(
      cuf, ntr, thrs, bsp, ws_score, ws_tok, ws_pos, (int*)d_out, B);
}